// GraphEncoder_13228499272011
// MI455X (gfx1250) — compile-verified
//
#include <hip/hip_runtime.h>
#include <hip/hip_bf16.h>

typedef __attribute__((ext_vector_type(2))) float v2f;
typedef __attribute__((ext_vector_type(8))) float v8f;

// ---------------------------------------------------------------------------
// Degree / normalization
// ---------------------------------------------------------------------------
__global__ void k_init_deg(float* __restrict__ deg, int n) {
    int i = blockIdx.x * blockDim.x + threadIdx.x;
    if (i < n) deg[i] = 1.0f;  // self loop
}

__global__ void k_deg_edges(const int* __restrict__ dst, float* __restrict__ deg, int E) {
    int e = blockIdx.x * blockDim.x + threadIdx.x;
    if (e < E) atomicAdd(&deg[dst[e]], 1.0f);
}

__global__ void k_dis(float* __restrict__ deg, int n) {
    int i = blockIdx.x * blockDim.x + threadIdx.x;
    if (i < n) {
        float d = deg[i];
        deg[i] = (d > 0.0f) ? rsqrtf(d) : 0.0f;  // d >= 1 always (self loops)
    }
}

// ---------------------------------------------------------------------------
// Dense GEMM via V_WMMA_F32_16X16X4_F32.  out[n,64] = A[n,K] @ W[K,64].
// K must be a multiple of 4 (here 4 or 64).  One wave -> 16x64 output tile.
// A-frag (16x4 f32): lane L holds row (L&15), K-pair base 2*(L>>4).
// B-frag (4x16 f32): lane L holds col (L&15), K-pair base 2*(L>>4).
// C/D   (16x16 f32): VGPR r, lane L -> row r + 8*(L>>4), col (L&15).
// ---------------------------------------------------------------------------
__global__ void k_gemm_wmma(const float* __restrict__ A, const float* __restrict__ W,
                            float* __restrict__ out, int n, int K) {
    __shared__ float sW[64 * 64];  // 16 KB max (K<=64), WGP has 320 KB LDS
    const int tid = threadIdx.x;   // 128 threads = 4 waves (wave32)
    for (int i = tid; i < K * 64; i += 128) sW[i] = W[i];
    __syncthreads();

    const int wave = tid >> 5;
    const int lane = tid & 31;
    const int tile_row = blockIdx.x * 64 + wave * 16;

    const int m    = lane & 15;        // A row within tile / D col
    const int koff = (lane >> 4) * 2;  // K-pair base for this half-wave
    int row = tile_row + m;
    if (row >= n) row = n - 1;         // clamp loads; stores guarded (EXEC stays all-1s)

    v8f acc[4] = {v8f{0.f}, v8f{0.f}, v8f{0.f}, v8f{0.f}};

    for (int kb = 0; kb < K; kb += 4) {
        v2f a;
        const float* ap = A + (size_t)row * K + kb + koff;
        a.x = ap[0];
        a.y = ap[1];
#pragma unroll
        for (int nt = 0; nt < 4; ++nt) {
            v2f b;
            const int col = nt * 16 + (lane & 15);
            b.x = sW[(kb + koff) * 64 + col];
            b.y = sW[(kb + koff + 1) * 64 + col];
            acc[nt] = __builtin_amdgcn_wmma_f32_16x16x4_f32(
                /*neg_a=*/false, a, /*neg_b=*/false, b,
                /*c_mod=*/(short)0, acc[nt], /*reuse_a=*/false, /*reuse_b=*/false);
        }
    }

#pragma unroll
    for (int nt = 0; nt < 4; ++nt) {
#pragma unroll
        for (int r = 0; r < 8; ++r) {
            const int mm = r + 8 * (lane >> 4);
            const int rr = tile_row + mm;
            if (rr < n) out[(size_t)rr * 64 + nt * 16 + (lane & 15)] = acc[nt][r];
        }
    }
}

// ---------------------------------------------------------------------------
// acc[i,:] = dis[i]^2 * h[i,:]   (self-loop contribution; initializes acc)
// ---------------------------------------------------------------------------
__global__ void k_self_init(const float* __restrict__ dis, const float* __restrict__ h,
                            float* __restrict__ acc, int n) {
    int t = blockIdx.x * blockDim.x + threadIdx.x;  // n*16 threads, 4 floats each
    if (t >= n * 16) return;
    int i = t >> 4;
    int jj = (t & 15) * 4;
    float di = dis[i];
    float s = di * di;
    const float4 hv = *(const float4*)&h[(size_t)i * 64 + jj];
    float4 o;
    o.x = s * hv.x; o.y = s * hv.y; o.z = s * hv.z; o.w = s * hv.w;
    *(float4*)&acc[(size_t)i * 64 + jj] = o;
}

// ---------------------------------------------------------------------------
// Edge scatter: acc[dst,:] += dis[src]*dis[dst] * h[src,:]
// 16 threads/edge, float4 gather + 4 atomic f32 adds (L2-resident on MI455X)
// ---------------------------------------------------------------------------
__global__ void k_scatter(const int* __restrict__ src, const int* __restrict__ dst,
                          const float* __restrict__ dis, const float* __restrict__ h,
                          float* __restrict__ acc, int E) {
    int t = blockIdx.x * blockDim.x + threadIdx.x;  // E*16 threads
    if (t >= E * 16) return;
    int e = t >> 4;
    int jj = (t & 15) * 4;
    int s = src[e];
    int d = dst[e];
    float norm = dis[s] * dis[d];
    const float4 hv = *(const float4*)&h[(size_t)s * 64 + jj];
    float* ap = &acc[(size_t)d * 64 + jj];
    atomicAdd(ap + 0, norm * hv.x);
    atomicAdd(ap + 1, norm * hv.y);
    atomicAdd(ap + 2, norm * hv.z);
    atomicAdd(ap + 3, norm * hv.w);
}

// acc[i,j] = relu(acc[i,j] + b[j])  in place
__global__ void k_bias_relu(float* __restrict__ acc, const float* __restrict__ b, int n) {
    int t = blockIdx.x * blockDim.x + threadIdx.x;  // n*64 threads
    if (t >= n * 64) return;
    int j = t & 63;
    acc[t] = fmaxf(acc[t] + b[j], 0.0f);
}

__global__ void k_zero(float* __restrict__ p, int n) {
    int t = blockIdx.x * blockDim.x + threadIdx.x;
    if (t < n) p[t] = 0.0f;
}

// sums[batch[i], :] += relu(acc2[i,:] + b2[:])
__global__ void k_pool(const float* __restrict__ acc2, const float* __restrict__ b2,
                       const int* __restrict__ batch, float* __restrict__ sums, int n) {
    int t = blockIdx.x * blockDim.x + threadIdx.x;  // n*16 threads
    if (t >= n * 16) return;
    int i = t >> 4;
    int jj = (t & 15) * 4;
    int g = batch[i];
    const float4 hv = *(const float4*)&acc2[(size_t)i * 64 + jj];
    const float4 bv = *(const float4*)&b2[jj];
    float* sp = &sums[(size_t)g * 64 + jj];
    atomicAdd(sp + 0, fmaxf(hv.x + bv.x, 0.0f));
    atomicAdd(sp + 1, fmaxf(hv.y + bv.y, 0.0f));
    atomicAdd(sp + 2, fmaxf(hv.z + bv.z, 0.0f));
    atomicAdd(sp + 3, fmaxf(hv.w + bv.w, 0.0f));
}

__global__ void k_count(const int* __restrict__ batch, float* __restrict__ cnt, int n) {
    int i = blockIdx.x * blockDim.x + threadIdx.x;
    if (i < n) atomicAdd(&cnt[batch[i]], 1.0f);
}

__global__ void k_div(const float* __restrict__ sums, const float* __restrict__ cnt,
                      float* __restrict__ out, int G) {
    int t = blockIdx.x * blockDim.x + threadIdx.x;  // G*64 threads
    if (t >= G * 64) return;
    out[t] = sums[t] / fmaxf(cnt[t >> 6], 1.0f);
}

// ---------------------------------------------------------------------------
extern "C" void kernel_launch(void* const* d_in, const int* in_sizes, int n_in,
                              void* d_out, int out_size, void* d_ws, size_t ws_size,
                              hipStream_t stream) {
    const float* x     = (const float*)d_in[0];
    const int* eidx    = (const int*)d_in[1];
    const int* batch   = (const int*)d_in[2];
    // d_in[3] = num_graphs scalar (device); use out_size instead (host-known)
    const float* W1 = (const float*)d_in[4];
    const float* b1 = (const float*)d_in[5];
    const float* W2 = (const float*)d_in[6];
    const float* b2 = (const float*)d_in[7];

    const int n = in_sizes[0] / 4;   // nodes
    const int E = in_sizes[1] / 2;   // edges
    const int G = out_size / 64;     // graphs

    const int* src = eidx;
    const int* dst = eidx + E;

    // Workspace layout (floats)
    float* ws   = (float*)d_ws;
    float* dis  = ws;                          // [n]     degree -> rsqrt(degree)
    float* bufA = dis + n;                     // [n*64]  h1 (pre-agg), then h2 (pre-agg)
    float* bufB = bufA + (size_t)n * 64;       // [n*64]  acc1 -> h1r, then acc2
    float* sums = bufB + (size_t)n * 64;       // [G*64]
    float* cnt  = sums + (size_t)G * 64;       // [G]

    const int T = 256;
    auto blk = [T](int work) { return (work + T - 1) / T; };

    // normalization coefficients
    k_init_deg<<<blk(n), T, 0, stream>>>(dis, n);
    k_deg_edges<<<blk(E), T, 0, stream>>>(dst, dis, E);
    k_dis<<<blk(n), T, 0, stream>>>(dis, n);

    // layer 1: h1 = x @ W1 (WMMA f32, K=4), aggregate, +b1, relu
    k_gemm_wmma<<<(n + 63) / 64, 128, 0, stream>>>(x, W1, bufA, n, 4);
    k_self_init<<<blk(n * 16), T, 0, stream>>>(dis, bufA, bufB, n);
    k_scatter<<<blk(E * 16), T, 0, stream>>>(src, dst, dis, bufA, bufB, E);
    k_bias_relu<<<blk(n * 64), T, 0, stream>>>(bufB, b1, n);

    // layer 2: h2 = h1r @ W2 (WMMA f32, K=64), aggregate
    k_gemm_wmma<<<(n + 63) / 64, 128, 0, stream>>>(bufB, W2, bufA, n, 64);
    k_self_init<<<blk(n * 16), T, 0, stream>>>(dis, bufA, bufB, n);
    k_scatter<<<blk(E * 16), T, 0, stream>>>(src, dst, dis, bufA, bufB, E);

    // global mean pool (bias + relu fused into pooling accumulation)
    k_zero<<<blk(G * 64 + G), T, 0, stream>>>(sums, G * 64 + G);
    k_pool<<<blk(n * 16), T, 0, stream>>>(bufB, b2, batch, sums, n);
    k_count<<<blk(n), T, 0, stream>>>(batch, cnt, n);
    k_div<<<blk(G * 64), T, 0, stream>>>(sums, cnt, (float*)d_out, G);
}